// DistributionLoss_15728170238686
// MI455X (gfx1250) — compile-verified
//
#include <hip/hip_runtime.h>
#include <hip/hip_bf16.h>
#include <math.h>

typedef float v2f __attribute__((ext_vector_type(2)));
typedef float v8f __attribute__((ext_vector_type(8)));

#define NBLK 1024
#define BSZ  256
#define NQ   32      // per-block partials: [0]=bce, [1]=n_valid, [2+bin*3+{0,1,2}]=cnt,sum_p,sum_t
#define NBINQ 30     // 10 bins * 3 quantities

// Wave-wide f32 sum using V_WMMA_F32_16X16X4_F32 with B == ones.
// A layout (documented): lane m -> A[m][0],A[m][1]; lane m+16 -> A[m][2],A[m][3].
// With a = {v, 0}, D[m][*] = v[m] + v[m+16].  C/D layout (documented): lane n holds
// rows m = n<16 ? 0..7 : 8..15 in c[0..7].  Sum those 8, then fold halves with shfl_xor(16).
// Requires full EXEC (all call sites are convergent).
__device__ __forceinline__ float wave_sum32(float v) {
    v2f a; a[0] = v;    a[1] = 0.0f;
    v2f b; b[0] = 1.0f; b[1] = 1.0f;
    v8f c = {};
    c = __builtin_amdgcn_wmma_f32_16x16x4_f32(false, a, false, b, (short)0, c, false, false);
    float s = ((c[0] + c[1]) + (c[2] + c[3])) + ((c[4] + c[5]) + (c[6] + c[7]));
    s += __shfl_xor(s, 16, 32);
    return s;   // full 32-lane sum in every lane
}

__device__ __forceinline__ void process_elem(float x, float t_raw,
                                             float& bce_acc, float& w_acc,
                                             float* lds_bins, int tid) {
    bool valid = (x == x) && (t_raw == t_raw);
    float w = valid ? 1.0f : 0.0f;
    x = valid ? x : 0.0f;
    float t = valid ? fminf(fmaxf(t_raw, 0.0f), 1.0f) : 0.0f;

    float ax  = fabsf(x);
    float e   = __expf(-ax);          // v_exp_f32 (trans)
    float opE = 1.0f + e;
    float sp  = fmaxf(-x, 0.0f) + __logf(opE);            // softplus(-x), stable
    float p   = (x >= 0.0f ? 1.0f : e) * __builtin_amdgcn_rcpf(opE);  // sigmoid(x)

    float a1   = 1.0f - t;
    float coef = fmaf(79.0f, t, 1.0f);                    // 80*t + (1-t)
    float u    = fmaf(coef, sp, a1 * x);                  // per-element BCE
    bce_acc    = fmaf(w, u, bce_acc);
    w_acc     += w;

    int bin = (int)(p * 10.0f);                           // trunc == floor (p >= 0)
    bin = min(bin, 9);
    int base = bin * (3 * BSZ) + tid;                     // bank = tid mod 64 -> conflict-free
    atomicAdd(&lds_bins[base],           w);              // ds_add_f32 (thread-private slot)
    atomicAdd(&lds_bins[base + BSZ],     p * w);
    atomicAdd(&lds_bins[base + 2 * BSZ], t * w);
}

__global__ __launch_bounds__(BSZ) void k_partial(const float* __restrict__ pred,
                                                 const float* __restrict__ tgt,
                                                 float* __restrict__ ws, int n) {
    __shared__ float lds_bins[NBINQ * BSZ];   // 30 KiB
    __shared__ float lds_red[2 * 8];

    const int tid  = threadIdx.x;
    const int lane = tid & 31;
    const int wav  = tid >> 5;

    for (int i = tid; i < NBINQ * BSZ; i += BSZ) lds_bins[i] = 0.0f;
    __syncthreads();

    float bce_acc = 0.0f, w_acc = 0.0f;

    const int n4     = n >> 2;
    const int stride = gridDim.x * blockDim.x;
    const float4* p4 = (const float4*)pred;
    const float4* t4 = (const float4*)tgt;

    for (int i = blockIdx.x * blockDim.x + tid; i < n4; i += stride) {
        // L2 streaming prefetch: one lane per 128B line, 4 grid-strides ahead (~32MB window in 192MB L2)
        if ((tid & 7) == 0 && (i + 4 * stride) < n4) {
            __builtin_prefetch(&p4[i + 4 * stride], 0, 1);   // global_prefetch_b8
            __builtin_prefetch(&t4[i + 4 * stride], 0, 1);
        }
        float4 xv = p4[i];
        float4 tv = t4[i];
        process_elem(xv.x, tv.x, bce_acc, w_acc, lds_bins, tid);
        process_elem(xv.y, tv.y, bce_acc, w_acc, lds_bins, tid);
        process_elem(xv.z, tv.z, bce_acc, w_acc, lds_bins, tid);
        process_elem(xv.w, tv.w, bce_acc, w_acc, lds_bins, tid);
    }
    // scalar tail (n % 4), handled once by block 0 / thread 0
    if (blockIdx.x == 0 && tid == 0) {
        for (int i = n4 << 2; i < n; ++i)
            process_elem(pred[i], tgt[i], bce_acc, w_acc, lds_bins, tid);
    }
    __syncthreads();   // all ds_add_f32 complete & visible

    // --- block reduction, WMMA-assisted wave sums (full EXEC, wave-uniform branches) ---
    float bs = wave_sum32(bce_acc);
    float vs = wave_sum32(w_acc);
    if (lane == 0) { lds_red[wav] = bs; lds_red[8 + wav] = vs; }

    float* wsb = ws + (size_t)blockIdx.x * NQ;
#pragma unroll
    for (int r = 0; r < 4; ++r) {
        int q = wav + 8 * r;            // wave-uniform
        if (q < NBINQ) {
            float s = 0.0f;
#pragma unroll
            for (int j = 0; j < BSZ / 32; ++j) s += lds_bins[q * BSZ + lane + 32 * j];
            s = wave_sum32(s);
            if (lane == 0) wsb[2 + q] = s;
        }
    }
    __syncthreads();
    if (tid == 0) {
        float b = 0.0f, v = 0.0f;
#pragma unroll
        for (int i = 0; i < 8; ++i) { b += lds_red[i]; v += lds_red[8 + i]; }
        wsb[0] = b;
        wsb[1] = v;
    }
}

__global__ __launch_bounds__(1024) void k_final(const float* __restrict__ ws,
                                                float* __restrict__ out, int nblocks) {
    __shared__ double tot[NQ];
    const int tid  = threadIdx.x;
    const int q    = tid >> 5;          // one wave per quantity
    const int lane = tid & 31;

    double acc = 0.0;
    for (int b = lane; b < nblocks; b += 32) acc += (double)ws[(size_t)b * NQ + q];
#pragma unroll
    for (int off = 16; off; off >>= 1) acc += __shfl_xor(acc, off, 32);
    if (lane == 0) tot[q] = acc;
    __syncthreads();

    if (tid == 0) {
        double bce_mean = tot[0] / tot[1];
        double dist = 0.0;
        for (int b = 0; b < 10; ++b) {
            double c    = tot[2 + b * 3 + 0];
            double sp   = tot[2 + b * 3 + 1];
            double st   = tot[2 + b * 3 + 2];
            double safe = c > 1.0 ? c : 1.0;
            double diff = fabs(sp / safe - st / safe);
            if (c > 10.0) dist += diff;
        }
        out[0] = (float)(bce_mean + dist / 10.0 * 0.2);
    }
}

extern "C" void kernel_launch(void* const* d_in, const int* in_sizes, int n_in,
                              void* d_out, int out_size, void* d_ws, size_t ws_size,
                              hipStream_t stream) {
    const float* pred = (const float*)d_in[0];
    const float* tgt  = (const float*)d_in[1];
    const int n = in_sizes[0];

    int nblocks = NBLK;
    size_t need = (size_t)nblocks * NQ * sizeof(float);
    if (ws_size < need) {
        nblocks = (int)(ws_size / (NQ * sizeof(float)));
        if (nblocks < 1) nblocks = 1;
    }
    float* ws = (float*)d_ws;

    k_partial<<<nblocks, BSZ, 0, stream>>>(pred, tgt, ws, n);
    k_final<<<1, 1024, 0, stream>>>(ws, (float*)d_out, nblocks);
}